// Clash_73718818668636
// MI455X (gfx1250) — compile-verified
//
#include <hip/hip_runtime.h>
#include <hip/hip_bf16.h>

typedef float v2f __attribute__((ext_vector_type(2)));
typedef float v8f __attribute__((ext_vector_type(8)));

#define NC 8                      // N_CHAINS
#define THR2 (1.1f * 1.1f)        // af3_clash_threshold^2
#define COUNT_THR 100.0f
#define REL_THR 0.5f

// ---------------------------------------------------------------------------
// Kernel 0: zero the scratch accumulators (counts + per-chain atom counts)
// ---------------------------------------------------------------------------
__global__ void zero_ws_kernel(unsigned int* __restrict__ gcounts,
                               int* __restrict__ nper, int S) {
    int t = blockIdx.x * blockDim.x + threadIdx.x;
    if (t < S * NC * NC) gcounts[t] = 0u;
    if (t < NC) nper[t] = 0;
}

// ---------------------------------------------------------------------------
// Kernel 1: per-atom chain id = asym_id[atom_to_token_idx[i]], and n_per_chain
// ---------------------------------------------------------------------------
__global__ void build_chain_kernel(const int* __restrict__ asym_id,
                                   const int* __restrict__ a2t,
                                   int* __restrict__ chain,
                                   int* __restrict__ nper, int nAtoms) {
    int i = blockIdx.x * blockDim.x + threadIdx.x;
    if (i < nAtoms) {
        int ch = asym_id[a2t[i]];
        chain[i] = ch;
        atomicAdd(&nper[ch], 1);
    }
}

// ---------------------------------------------------------------------------
// Kernel 2: WMMA pair-tile kernel.
// One wave owns a 16-row i-tile and sweeps 8 16-col j-tiles.
// d2 tile = A(16x4) * B(4x16) + C via v_wmma_f32_16x16x4_f32:
//   A row i = (-2x_i, -2y_i, -2z_i, 1)
//   B col j = ( x_j,   y_j,   z_j, sq_j)
//   C_ij    = sq_i
// => D_ij = sq_i + sq_j - 2*(x_i . x_j)   (exact reference formulation, f32)
//
// Epilogue: build an 8-bit per-lane clash mask, then branch over the whole
// atomic block (s_cbranch_execz fast path — clashes are ~0.1% dense).
// ---------------------------------------------------------------------------
__global__ __launch_bounds__(256) void clash_tiles_kernel(
    const float* __restrict__ X,        // [S, nAtoms, 3]
    const int* __restrict__ chain,      // [nAtoms]
    unsigned int* __restrict__ gcounts, // [S, NC, NC]
    int nAtoms) {
    __shared__ unsigned int cnt[NC * NC];

    const int t = threadIdx.x;
    if (t < NC * NC) cnt[t] = 0u;
    __syncthreads();

    const int lane = t & 31;
    const int wave = t >> 5;
    const bool hi  = (lane >= 16);      // upper half-wave: K=2,3 / M+8 rows
    const int s    = blockIdx.z;
    const int i0   = blockIdx.x * 16;                 // row tile base
    const int jb   = (blockIdx.y * 8 + wave) * 128;   // this wave's 128-col span

    const float* __restrict__ Xs = X + (size_t)s * (size_t)nAtoms * 3;

    // ---- A fragment (16x4 f32: lanes 0-15 -> K=0,1; lanes 16-31 -> K=2,3) ----
    const int rowA = i0 + (lane & 15);
    const float ax = Xs[rowA * 3 + 0];
    const float ay = Xs[rowA * 3 + 1];
    const float az = Xs[rowA * 3 + 2];
    v2f a;
    a.x = hi ? (-2.0f * az) : (-2.0f * ax);
    a.y = hi ? 1.0f         : (-2.0f * ay);

    // ---- C fragment (sq_i per row) + row chain bin bases, loaded once ----
    const int rbase = i0 + (hi ? 8 : 0);
    v8f c;
    int ciNC[8];                         // chain[i] * NC, hoisted out of j loop
#pragma unroll
    for (int v = 0; v < 8; ++v) {
        const int r = rbase + v;
        const float x = Xs[r * 3 + 0];
        const float y = Xs[r * 3 + 1];
        const float z = Xs[r * 3 + 2];
        c[v] = fmaf(x, x, fmaf(y, y, z * z));
        ciNC[v] = chain[r] * NC;
    }

    // ---- sweep 8 j-tiles, one WMMA each ----
#pragma unroll
    for (int jt = 0; jt < 8; ++jt) {
        const int j = jb + jt * 16 + (lane & 15);
        const float bx = Xs[j * 3 + 0];
        const float by = Xs[j * 3 + 1];
        const float bz = Xs[j * 3 + 2];
        const float sqj = fmaf(bx, bx, fmaf(by, by, bz * bz));
        const int cj = chain[j];
        const int cjNC = cj * NC;

        v2f b;
        b.x = hi ? bz  : bx;
        b.y = hi ? sqj : by;

        // (neg_a, A, neg_b, B, c_mod, C, reuse_a, reuse_b)
        v8f d = __builtin_amdgcn_wmma_f32_16x16x4_f32(
            false, a, false, b, (short)0, c, false, false);

        // per-lane clash mask; i == j lands on ci == cj (chain diag excluded)
        unsigned mask = 0u;
#pragma unroll
        for (int v = 0; v < 8; ++v)
            mask |= ((d[v] < THR2) && (ciNC[v] != cjNC)) ? (1u << v) : 0u;

        if (mask) {                      // wave-level s_cbranch_execz fast path
#pragma unroll
            for (int v = 0; v < 8; ++v)
                if (mask & (1u << v))
                    atomicAdd(&cnt[ciNC[v] + cj], 1u);
        }
    }

    __syncthreads();
    if (t < NC * NC && cnt[t] != 0u)
        atomicAdd(&gcounts[(size_t)s * NC * NC + t], cnt[t]);
}

// ---------------------------------------------------------------------------
// Kernel 3: finalize. d_out layout (return order, flat):
//   [0 .. S*64)              has_af3_clash as 0/1 float
//   [S*64 .. S*64 + S*128)   details: [S][a][b][{total, rel}]
// ---------------------------------------------------------------------------
__global__ void finalize_kernel(const unsigned int* __restrict__ gcounts,
                                const int* __restrict__ nper,
                                float* __restrict__ out, int S) {
    int t = blockIdx.x * blockDim.x + threadIdx.x;
    if (t >= S * NC * NC) return;
    const int ab = t & (NC * NC - 1);
    const int aa = ab >> 3;
    const int bb = ab & 7;

    float total = (aa == bb) ? 0.0f : (float)gcounts[t];
    float mn = (float)min(nper[aa], nper[bb]);
    float rel = total / mn;                        // matches reference (inf/nan ok)
    bool has = (total > COUNT_THR) || (rel > REL_THR);

    out[t] = has ? 1.0f : 0.0f;
    out[S * NC * NC + 2 * t + 0] = total;
    out[S * NC * NC + 2 * t + 1] = rel;
}

// ---------------------------------------------------------------------------
extern "C" void kernel_launch(void* const* d_in, const int* in_sizes, int n_in,
                              void* d_out, int out_size, void* d_ws, size_t ws_size,
                              hipStream_t stream) {
    const float* X      = (const float*)d_in[0];  // pred_coordinate [S, N, 3]
    const int* asym_id  = (const int*)d_in[1];    // [N_TOKENS]
    const int* a2t      = (const int*)d_in[2];    // [N_ATOMS]

    const int nAtoms = in_sizes[2];               // 8192
    const int S      = in_sizes[0] / (nAtoms * 3);// 4

    // workspace layout: chain[nAtoms] | nper[NC] | gcounts[S*NC*NC]
    int* chain            = (int*)d_ws;
    int* nper             = chain + nAtoms;
    unsigned int* gcounts = (unsigned int*)(nper + NC);

    zero_ws_kernel<<<1, 256, 0, stream>>>(gcounts, nper, S);
    build_chain_kernel<<<(nAtoms + 255) / 256, 256, 0, stream>>>(
        asym_id, a2t, chain, nper, nAtoms);

    // grid: 512 i-tiles x 8 j-chunks x S samples; 8 waves/block, 8 WMMAs/wave
    dim3 grid(nAtoms / 16, nAtoms / 1024, S);
    clash_tiles_kernel<<<grid, 256, 0, stream>>>(X, chain, gcounts, nAtoms);

    finalize_kernel<<<1, 256, 0, stream>>>(gcounts, nper, (float*)d_out, S);
}